// EdgeConv_16114717294920
// MI455X (gfx1250) — compile-verified
//
#include <hip/hip_runtime.h>
#include <math.h>

typedef __attribute__((ext_vector_type(2))) float v2f;
typedef __attribute__((ext_vector_type(8))) float v8f;

#define WMMA_F32(a, b, c) \
    __builtin_amdgcn_wmma_f32_16x16x4_f32(false, (a), false, (b), (short)0, (c), false, false)

#define BN_EPS 1e-3f
#define BIG 3.0e38f

// ---- GEMM over a 16 x 64 A tile (row-major, stride 64 floats in LDS) times a
// ---- transposed 64 x 64 weight tile Wt (Wt[d*64 + c] = W[c][d]), 4 N-tiles.
__device__ __forceinline__ void gemm16x64(const float* A, const float* Wt,
                                          int lrow, int half, v8f acc[4]) {
#pragma unroll
    for (int kk = 0; kk < 16; ++kk) {
        const int cb = (kk << 2) + (half << 1);
        v2f a = *(const v2f*)(A + (lrow << 6) + cb);
#pragma unroll
        for (int t = 0; t < 4; ++t) {
            v2f bw = *(const v2f*)(Wt + ((lrow + (t << 4)) << 6) + cb);
            acc[t] = WMMA_F32(a, bw, acc[t]);
        }
    }
}

__device__ __forceinline__ void bn_relu(v8f acc[4], const float* g, const float* be,
                                        const float* mn, const float* vr, int lrow) {
#pragma unroll
    for (int t = 0; t < 4; ++t) {
        const int d = lrow + (t << 4);
        const float s = g[d] * (1.0f / sqrtf(vr[d] + BN_EPS));
        const float bi = be[d] - mn[d] * s;
#pragma unroll
        for (int v = 0; v < 8; ++v) acc[t][v] = fmaxf(acc[t][v] * s + bi, 0.0f);
    }
}

__device__ __forceinline__ void store16x64(const v8f acc[4], int lrow, int half, float* Y) {
#pragma unroll
    for (int t = 0; t < 4; ++t)
#pragma unroll
        for (int v = 0; v < 8; ++v)
            Y[((v + (half << 3)) << 6) + lrow + (t << 4)] = acc[t][v];
}

__global__ __launch_bounds__(256) void edgeconv_fused_kernel(
    const float* __restrict__ points,  // (B,P,3)
    const float* __restrict__ feats,   // (B,P,64)
    const float* __restrict__ w0,      // (128,64)
    const float* __restrict__ w1,      // (64,64)
    const float* __restrict__ w2,      // (64,64)
    const float* __restrict__ gammas,  // (3,64)
    const float* __restrict__ betas,   // (3,64)
    const float* __restrict__ means,   // (3,64)
    const float* __restrict__ vars,    // (3,64)
    const float* __restrict__ scw,     // (64,64)
    const float* __restrict__ scg, const float* __restrict__ scb,
    const float* __restrict__ scm, const float* __restrict__ scv,
    float* __restrict__ out)           // (B,P,64)
{
    extern __shared__ float smem[];
    // LDS carve (floats)
    float* pts     = smem;               // 768   : batch points
    float* Wt0     = pts + 768;          // 8192  : w0 transposed [64][128]
    float* Wt1     = Wt0 + 8192;         // 4096  : w1 transposed [64][64]
    float* Wt2     = Wt1 + 4096;         // 4096
    float* scWt    = Wt2 + 4096;         // 4096
    float* waveBuf = scWt + 4096;        // 8 * 2048 (two 16x64 tiles per wave)
    float* ctrAll  = waveBuf + 16384;    // 8 * 64
    float* ftsAll  = ctrAll + 512;       // 8 * 64
    int*   nbrAll  = (int*)(ftsAll + 512); // 8 * 16

    const int tid  = threadIdx.x;
    const int lane = tid & 31;
    const int w    = tid >> 5;
    const int lrow = lane & 15;
    const int half = lane >> 4;
    const int b    = blockIdx.x >> 5;
    const int pi   = ((blockIdx.x & 31) << 3) | w;

    // ---- cooperative block loads: points + transposed weights ----
    for (int i = tid; i < 768; i += 256) pts[i] = points[b * 768 + i];
    for (int i = tid; i < 8192; i += 256) {
        const int d = i >> 7, c = i & 127;
        Wt0[i] = w0[c * 64 + d];
    }
    for (int i = tid; i < 4096; i += 256) {
        const int d = i >> 6, c = i & 63;
        Wt1[i]  = w1[c * 64 + d];
        Wt2[i]  = w2[c * 64 + d];
        scWt[i] = scw[c * 64 + d];
    }
    __syncthreads();

    float* bufA = waveBuf + (w << 11);   // edge-X, later L1 output
    float* bufB = bufA + 1024;           // L0 output
    float* cC   = ctrAll + (w << 6);
    float* ftsW = ftsAll + (w << 6);
    int*   nbrW = nbrAll + (w << 4);

    // =================== KNN (16 nearest, excluding self) ===================
    const float px = pts[pi * 3], py = pts[pi * 3 + 1], pz = pts[pi * 3 + 2];
    float dist[8];
#pragma unroll
    for (int j = 0; j < 8; ++j) {
        const int q = lane + (j << 5);
        const float dx = pts[q * 3] - px;
        const float dy = pts[q * 3 + 1] - py;
        const float dz = pts[q * 3 + 2] - pz;
        const float dd = dx * dx + dy * dy + dz * dz;
        dist[j] = (q == pi) ? BIG : dd;
    }
    for (int it = 0; it < 16; ++it) {
        float bv = dist[0];
        int bj = 0;
#pragma unroll
        for (int j = 1; j < 8; ++j)
            if (dist[j] < bv) { bv = dist[j]; bj = j; }
        int bq = lane + (bj << 5);
#pragma unroll
        for (int off = 16; off >= 1; off >>= 1) {
            const float ov = __shfl_xor(bv, off, 32);
            const int   oq = __shfl_xor(bq, off, 32);
            if (ov < bv || (ov == bv && oq < bq)) { bv = ov; bq = oq; }
        }
        if ((bq & 31) == lane) {
            const int jw = bq >> 5;
#pragma unroll
            for (int j = 0; j < 8; ++j)
                if (j == jw) dist[j] = BIG;
        }
        if (lane == 0) nbrW[it] = bq;
    }

    // =================== build center vector + edge matrix ===================
    const float* fbase = feats + ((b << 8) | pi) * 64;
    const float cf0 = fbase[lane];
    const float cf1 = fbase[lane + 32];
    cC[lane]      = cf0;
    cC[lane + 32] = cf1;
    for (int r = 0; r < 16; ++r) {
        const int nq = nbrW[r];
        const float* nf = feats + (((b << 8) | nq)) * 64;
        bufA[(r << 6) + lane]      = nf[lane]      - cf0;
        bufA[(r << 6) + lane + 32] = nf[lane + 32] - cf1;
    }

    // =================== layer 0: [center, edge](16x128) @ w0(128x64) ===================
    const v8f vz = {0.f, 0.f, 0.f, 0.f, 0.f, 0.f, 0.f, 0.f};
    v8f acc[4] = {vz, vz, vz, vz};
    // center half: A fragment is row-independent broadcast of cC
#pragma unroll
    for (int kk = 0; kk < 16; ++kk) {
        const int cb = (kk << 2) + (half << 1);
        v2f a = *(const v2f*)(cC + cb);
#pragma unroll
        for (int t = 0; t < 4; ++t) {
            v2f bw = *(const v2f*)(Wt0 + (lrow + (t << 4)) * 128 + cb);
            acc[t] = WMMA_F32(a, bw, acc[t]);
        }
    }
    // edge half
#pragma unroll
    for (int kk = 0; kk < 16; ++kk) {
        const int cb = (kk << 2) + (half << 1);
        v2f a = *(const v2f*)(bufA + (lrow << 6) + cb);
#pragma unroll
        for (int t = 0; t < 4; ++t) {
            v2f bw = *(const v2f*)(Wt0 + (lrow + (t << 4)) * 128 + 64 + cb);
            acc[t] = WMMA_F32(a, bw, acc[t]);
        }
    }
    bn_relu(acc, gammas, betas, means, vars, lrow);
    store16x64(acc, lrow, half, bufB);

    // =================== layer 1 ===================
    acc[0] = vz; acc[1] = vz; acc[2] = vz; acc[3] = vz;
    gemm16x64(bufB, Wt1, lrow, half, acc);
    bn_relu(acc, gammas + 64, betas + 64, means + 64, vars + 64, lrow);
    store16x64(acc, lrow, half, bufA);

    // =================== layer 2 ===================
    acc[0] = vz; acc[1] = vz; acc[2] = vz; acc[3] = vz;
    gemm16x64(bufA, Wt2, lrow, half, acc);
    bn_relu(acc, gammas + 128, betas + 128, means + 128, vars + 128, lrow);

    // =================== mean over K (rows) ===================
#pragma unroll
    for (int t = 0; t < 4; ++t) {
        float s = acc[t][0] + acc[t][1] + acc[t][2] + acc[t][3] +
                  acc[t][4] + acc[t][5] + acc[t][6] + acc[t][7];
        s += __shfl_xor(s, 16, 32);
        if (lane < 16) ftsW[lrow + (t << 4)] = s * (1.0f / 16.0f);
    }

    // =================== shortcut GEMV + BN + add + ReLU ===================
    const int d0 = lane << 1;
    float s0 = 0.f, s1 = 0.f;
#pragma unroll
    for (int c = 0; c < 64; ++c) {
        const float f = cC[c];
        s0 += f * scWt[(d0 << 6) + c];
        s1 += f * scWt[((d0 + 1) << 6) + c];
    }
    const float sA = scg[d0] * (1.0f / sqrtf(scv[d0] + BN_EPS));
    const float bA = scb[d0] - scm[d0] * sA;
    const float sB = scg[d0 + 1] * (1.0f / sqrtf(scv[d0 + 1] + BN_EPS));
    const float bB = scb[d0 + 1] - scm[d0 + 1] * sB;
    v2f o;
    o.x = fmaxf(s0 * sA + bA + ftsW[d0], 0.f);
    o.y = fmaxf(s1 * sB + bB + ftsW[d0 + 1], 0.f);
    *(v2f*)(out + (((b << 8) | pi) << 6) + d0) = o;
}

extern "C" void kernel_launch(void* const* d_in, const int* in_sizes, int n_in,
                              void* d_out, int out_size, void* d_ws, size_t ws_size,
                              hipStream_t stream) {
    (void)in_sizes; (void)n_in; (void)d_ws; (void)ws_size; (void)out_size;
    const float* points = (const float*)d_in[0];
    const float* feats  = (const float*)d_in[1];
    const float* w0     = (const float*)d_in[2];
    const float* w1     = (const float*)d_in[3];
    const float* w2     = (const float*)d_in[4];
    const float* gammas = (const float*)d_in[5];
    const float* betas  = (const float*)d_in[6];
    const float* means  = (const float*)d_in[7];
    const float* vars   = (const float*)d_in[8];
    const float* scw    = (const float*)d_in[9];
    const float* scg    = (const float*)d_in[10];
    const float* scb    = (const float*)d_in[11];
    const float* scm    = (const float*)d_in[12];
    const float* scv    = (const float*)d_in[13];
    float* out = (float*)d_out;

    // LDS: 768 + 8192 + 3*4096 + 16384 + 512 + 512 floats + 128 ints
    const size_t ldsBytes = (768 + 8192 + 3 * 4096 + 16384 + 512 + 512) * sizeof(float)
                          + 128 * sizeof(int);
    dim3 grid(4096), block(256);
    edgeconv_fused_kernel<<<grid, block, ldsBytes, stream>>>(
        points, feats, w0, w1, w2, gammas, betas, means, vars,
        scw, scg, scb, scm, scv, out);
}